// SoftDTW_18631568130929
// MI455X (gfx1250) — compile-verified
//
#include <hip/hip_runtime.h>

#define BIGV 10000000000.0f
#define NDIM 1024
#define TN 32      // tiles per dimension
#define TS 32      // tile size (rows == lanes)
#define DSTRIDE 36 // padded LDS row stride in floats (16B aligned, conflict-free)

__device__ __forceinline__ float softmin3(float a, float b, float c) {
    // matches jax logsumexp-with-max-subtraction exactly in structure
    float m = fminf(a, fminf(b, c));
    float s = __expf(m - a) + __expf(m - b) + __expf(m - c);
    return m - __logf(s);
}

typedef __attribute__((address_space(3))) float lds_f;

__device__ __forceinline__ void async_b128(unsigned lds_off, const float* g) {
    // per-lane: copies 16 bytes from global address g to LDS offset lds_off, tracked on ASYNCcnt
    asm volatile("global_load_async_to_lds_b128 %0, %1, off"
                 :: "v"(lds_off), "v"(g) : "memory");
}
__device__ __forceinline__ void wait_async0() {
    asm volatile("s_wait_asynccnt 0" ::: "memory");
}

__global__ __launch_bounds__(1024, 1)
void softdtw_kernel(const float* __restrict__ D, float* __restrict__ out) {
    // 2 x 32 waves x (32 rows x 36 floats) = 288 KB, + 12 KB boundary rows = 300 KB LDS
    __shared__ float dtile[2][TN][TS * DSTRIDE];
    __shared__ float bot[3][TN][TS];

    const int b    = blockIdx.x;
    const int w    = threadIdx.x >> 5;  // wave = tile row
    const int lane = threadIdx.x & 31;  // lane = row within tile

    const float* Db = D + ((size_t)b << 20);  // b * 1024 * 1024

    // b128 tile-load mapping: 8 issues; each issue, lane covers row 4k + lane/8,
    // 16-byte column group lane%8  (4 coalesced 128B rows per issue)
    const int lrow = lane >> 3;
    const int lcg  = lane & 7;

    // issue async loads for this wave's c = 0 tile immediately
    {
        const float* gbase = Db + (size_t)(w * TS) * NDIM;
#pragma unroll
        for (int k = 0; k < 8; ++k) {
            int r = 4 * k + lrow;
            unsigned lp = (unsigned)(size_t)(lds_f*)&dtile[0][w][r * DSTRIDE + lcg * 4];
            async_b128(lp, gbase + (size_t)r * NDIM + lcg * 4);
        }
    }

    // cur carries R[row][last computed col]; doubles as left-tile boundary.
    float cur = BIGV, cur_m1 = BIGV;

    for (int td = 0; td < 2 * TN - 1; ++td) {
        const int  c      = td - w;
        const bool active = (c >= 0) && (c < TN);
        if (active) {
            wait_async0();  // tile c resident in dtile[c&1]

            // prefetch tile c+1 into the other buffer (hidden behind compute)
            if (c + 1 < TN) {
                const float* gbase = Db + (size_t)(w * TS) * NDIM + (c + 1) * TS;
#pragma unroll
                for (int k = 0; k < 8; ++k) {
                    int r = 4 * k + lrow;
                    unsigned lp = (unsigned)(size_t)(lds_f*)
                        &dtile[(c + 1) & 1][w][r * DSTRIDE + lcg * 4];
                    async_b128(lp, gbase + (size_t)r * NDIM + lcg * 4);
                }
            }

            // boundary values: top row of tile above (written at td-1, buffer (td-1)%3)
            float topreg = BIGV;
            if (w > 0) topreg = bot[(td + 2) % 3][w - 1][lane];
            // corner: bottom-right of up-left tile (written at td-2, buffer (td-2)%3)
            float corner;
            if (w > 0 && c > 0)        corner = bot[(td + 1) % 3][w - 1][TS - 1];
            else if (w == 0 && c == 0) corner = 0.0f;   // R[-1][-1] = 0
            else                       corner = BIGV;

            const float* dt = dtile[c & 1][w];

            // 63 micro anti-diagonals; lane l computes cell (l, t-l)
            for (int t = 0; t < 2 * TS - 1; ++t) {
                float upv = __shfl_up(cur, 1);     // R[l-1][j]   (lane l-1, step t-1)
                float dgv = __shfl_up(cur_m1, 1);  // R[l-1][j-1] (lane l-1, step t-2)
                float top_t   = __shfl(topreg, t & 31);
                float top_tm1 = __shfl(topreg, (t - 1) & 31);
                if (lane == 0) {
                    upv = top_t;
                    dgv = (t == 0) ? corner : top_tm1;
                }
                if (t >= lane && t < lane + TS) {
                    int   j   = t - lane;
                    float dv  = dt[lane * DSTRIDE + j];
                    float val = dv + softmin3(dgv, upv, cur);  // diag, up, left
                    cur_m1 = cur;
                    cur    = val;
                    if (lane == TS - 1) bot[td % 3][w][j] = val;  // publish bottom row
                }
            }
        }
        __syncthreads();  // one barrier per tile-step
    }

    if (w == TN - 1 && lane == TS - 1) out[b] = cur;  // R[1023][1023]
}

extern "C" void kernel_launch(void* const* d_in, const int* in_sizes, int n_in,
                              void* d_out, int out_size, void* d_ws, size_t ws_size,
                              hipStream_t stream) {
    const float* D = (const float*)d_in[0];
    float* out = (float*)d_out;
    const int B = in_sizes[0] / (NDIM * NDIM);
    softdtw_kernel<<<B, TN * 32, 0, stream>>>(D, out);
}